// DifferentiableAddressingHead_37649683316887
// MI455X (gfx1250) — compile-verified
//
#include <hip/hip_runtime.h>
#include <hip/hip_bf16.h>

#define B_      512
#define M_      4096
#define D_      64
#define C_      256
#define EPS_    1e-8f
#define THREADS 256

typedef __attribute__((ext_vector_type(2))) float v2f;
typedef __attribute__((ext_vector_type(8))) float v8f;

__device__ __forceinline__ float blockSum(float v, float* red, int tid) {
    red[tid] = v;
    __syncthreads();
    #pragma unroll
    for (int s = 128; s > 0; s >>= 1) {
        if (tid < s) red[tid] = red[tid] + red[tid + s];
        __syncthreads();
    }
    float r = red[0];
    __syncthreads();
    return r;
}

__device__ __forceinline__ float blockMax(float v, float* red, int tid) {
    red[tid] = v;
    __syncthreads();
    #pragma unroll
    for (int s = 128; s > 0; s >>= 1) {
        if (tid < s) red[tid] = fmaxf(red[tid], red[tid + s]);
        __syncthreads();
    }
    float r = red[0];
    __syncthreads();
    return r;
}

__device__ __forceinline__ float softplusf(float x) {
    return (x > 20.f) ? x : log1pf(expf(x));
}

__global__ __launch_bounds__(THREADS)
void ntm_head_kernel(const float* __restrict__ memory,
                     const float* __restrict__ cstate,
                     const float* __restrict__ prevw,
                     const float* __restrict__ Wk,
                     const float* __restrict__ Wb,  const float* __restrict__ bb,
                     const float* __restrict__ Wgate, const float* __restrict__ bgate,
                     const float* __restrict__ Ws,  const float* __restrict__ bs,
                     const float* __restrict__ Wg,  const float* __restrict__ bg,
                     float* __restrict__ out) {
    __shared__ float arrA[M_];   // dots -> gated
    __shared__ float arrB[M_];   // sq-norms
    __shared__ float csL[C_];
    __shared__ float red[THREADS];
    __shared__ float qnL[D_];

    const int b   = blockIdx.x;
    const int tid = threadIdx.x;
    const float* mem_b  = memory + (size_t)b * M_ * D_;
    const float* cs_b   = cstate + (size_t)b * C_;
    const float* prev_b = prevw  + (size_t)b * M_;
    float*       out_b  = out    + (size_t)b * M_;

    // ---------------- Phase 0: controller projections (tiny) ----------------
    csL[tid] = cs_b[tid];
    __syncthreads();
    const float cst = csL[tid];

    float q = 0.f;
    if (tid < D_) {
        #pragma unroll 8
        for (int c = 0; c < C_; ++c) q = fmaf(csL[c], Wk[c * D_ + tid], q);
    }
    const float qn2     = blockSum((tid < D_) ? q * q : 0.f, red, tid);
    const float betaRaw = blockSum(cst * Wb[tid],       red, tid) + bb[0];
    const float gateRaw = blockSum(cst * Wgate[tid],    red, tid) + bgate[0];
    const float sRaw0   = blockSum(cst * Ws[tid * 3 + 0], red, tid) + bs[0];
    const float sRaw1   = blockSum(cst * Ws[tid * 3 + 1], red, tid) + bs[1];
    const float sRaw2   = blockSum(cst * Ws[tid * 3 + 2], red, tid) + bs[2];
    const float gamRaw  = blockSum(cst * Wg[tid],       red, tid) + bg[0];

    if (tid < D_) qnL[tid] = q / (sqrtf(qn2) + EPS_);
    __syncthreads();

    const float beta  = softplusf(betaRaw) + 1.f;
    const float gate  = 1.f / (1.f + expf(-gateRaw));
    const float gamma = softplusf(gamRaw) + 1.f;
    const float sm    = fmaxf(sRaw0, fmaxf(sRaw1, sRaw2));
    float s0 = expf(sRaw0 - sm), s1 = expf(sRaw1 - sm), s2 = expf(sRaw2 - sm);
    const float sInv = 1.f / (s0 + s1 + s2);
    s0 *= sInv; s1 *= sInv; s2 *= sInv;

    // ---------------- Phase 1: sim via WMMA f32 16x16x4 ----------------
    const int wave = tid >> 5;
    const int lane = tid & 31;
    const int rsel = lane & 15;          // row within 16-row tile
    const int koff = (lane >> 4) << 1;   // 0 for lanes 0-15, 2 for lanes 16-31

    // Pre-stage broadcast B fragments: B[k][n] = qn[k] for all n.
    v2f qf[16];
    #pragma unroll
    for (int kc = 0; kc < 16; ++kc) {
        qf[kc].x = qnL[kc * 4 + koff];
        qf[kc].y = qnL[kc * 4 + koff + 1];
    }

    for (int tile = wave; tile < M_ / 16; tile += 8) {
        const int m0 = tile * 16;
        const float* rowBase = mem_b + (size_t)(m0 + rsel) * D_ + koff;
        if (tile + 8 < M_ / 16)
            __builtin_prefetch(rowBase + 8 * 16 * D_, 0, 0);  // global_prefetch_b8

        v8f acc = {0.f, 0.f, 0.f, 0.f, 0.f, 0.f, 0.f, 0.f};
        float sq = 0.f;
        #pragma unroll
        for (int kc = 0; kc < 16; ++kc) {
            v2f a = *(const v2f*)(rowBase + kc * 4);   // global_load_b64
            sq = fmaf(a.x, a.x, sq);
            sq = fmaf(a.y, a.y, sq);
            // D = A(16x4 mem tile) * B(4x16 qn broadcast) + C
            acc = __builtin_amdgcn_wmma_f32_16x16x4_f32(
                false, a, false, qf[kc], (short)0, acc, false, false);
        }
        // full row ||.||^2 = this half + partner half (lane ^ 16)
        const float sqFull = sq + __shfl_xor(sq, 16, 32);
        if (lane < 16) arrB[m0 + lane] = sqFull;
        if (lane == 0) {
            #pragma unroll
            for (int v = 0; v < 8; ++v) arrA[m0 + v] = acc[v];       // rows 0-7
        }
        if (lane == 16) {
            #pragma unroll
            for (int v = 0; v < 8; ++v) arrA[m0 + 8 + v] = acc[v];   // rows 8-15
        }
    }
    __syncthreads();

    // ---------------- Phase 2: softmax(beta*sim) + gate ----------------
    float zv[16];
    float lmax = -3.402823466e38f;
    #pragma unroll
    for (int i = 0; i < 16; ++i) {
        const int m = tid + i * THREADS;
        const float z = beta * arrA[m] / (sqrtf(arrB[m]) + EPS_);
        zv[i] = z;
        lmax = fmaxf(lmax, z);
    }
    const float zmax = blockMax(lmax, red, tid);

    float lsum = 0.f;
    #pragma unroll
    for (int i = 0; i < 16; ++i) {
        const float e = expf(zv[i] - zmax);
        zv[i] = e;
        lsum += e;
    }
    const float denom = blockSum(lsum, red, tid);
    const float dInv  = 1.f / denom;

    #pragma unroll
    for (int i = 0; i < 16; ++i) {
        const int m = tid + i * THREADS;
        const float cw = zv[i] * dInv;
        arrA[m] = gate * cw + (1.f - gate) * prev_b[m];   // gated
    }
    __syncthreads();

    // ---------------- Phase 3: circular 3-tap shift + sharpen ----------------
    float psum = 0.f;
    #pragma unroll
    for (int i = 0; i < 16; ++i) {
        const int m   = tid + i * THREADS;
        const int mm1 = (m + M_ - 1) & (M_ - 1);
        const int mp1 = (m + 1) & (M_ - 1);
        const float shifted = s0 * arrA[mm1] + s1 * arrA[m] + s2 * arrA[mp1];
        const float sharp = expf(gamma * logf(shifted + EPS_));
        zv[i] = sharp;
        psum += sharp;
    }
    const float ssum = blockSum(psum, red, tid);
    const float oInv = 1.f / (ssum + EPS_);

    #pragma unroll
    for (int i = 0; i < 16; ++i) {
        const int m = tid + i * THREADS;
        out_b[m] = zv[i] * oInv;
    }
}

extern "C" void kernel_launch(void* const* d_in, const int* in_sizes, int n_in,
                              void* d_out, int out_size, void* d_ws, size_t ws_size,
                              hipStream_t stream) {
    const float* memory = (const float*)d_in[0];
    const float* cstate = (const float*)d_in[1];
    const float* prevw  = (const float*)d_in[2];
    const float* Wk     = (const float*)d_in[3];
    const float* Wb     = (const float*)d_in[4];
    const float* bb     = (const float*)d_in[5];
    const float* Wgate  = (const float*)d_in[6];
    const float* bgate  = (const float*)d_in[7];
    const float* Ws     = (const float*)d_in[8];
    const float* bs     = (const float*)d_in[9];
    const float* Wg     = (const float*)d_in[10];
    const float* bg     = (const float*)d_in[11];
    float* out = (float*)d_out;

    hipLaunchKernelGGL(ntm_head_kernel, dim3(B_), dim3(THREADS), 0, stream,
                       memory, cstate, prevw, Wk, Wb, bb, Wgate, bgate,
                       Ws, bs, Wg, bg, out);
}